// LogSumExp_3753801416753
// MI455X (gfx1250) — compile-verified
//
#include <hip/hip_runtime.h>

#define KP1 7            // coefficients for degrees 0..6 (K=5, P=2 -> KP=6)
#define NEGV (-1.0e30f)
#define TPB 256

// ---------------- CDNA5 async global->LDS staging (guarded) ----------------
#if defined(__gfx1250__) && __has_builtin(__builtin_amdgcn_global_load_async_to_lds_b128)
#define USE_ASYNC_LDS 1
typedef int v4i_t __attribute__((vector_size(4 * sizeof(int))));
typedef __attribute__((address_space(1))) v4i_t as1_v4i;
typedef __attribute__((address_space(3))) v4i_t as3_v4i;
__device__ __forceinline__ void async_copy_b128(const float* g, float* l) {
    __builtin_amdgcn_global_load_async_to_lds_b128((as1_v4i*)g, (as3_v4i*)l, 0, 0);
}
__device__ __forceinline__ void async_wait_all() {
#if __has_builtin(__builtin_amdgcn_s_wait_asynccnt)
    __builtin_amdgcn_s_wait_asynccnt(0);
#else
    asm volatile("s_wait_asynccnt 0" ::: "memory");
#endif
}
#else
#define USE_ASYNC_LDS 0
#endif

// ---------------- log-space helpers ----------------
__device__ __forceinline__ float laddexp(float a, float b) {
    float m = fmaxf(a, b);
    float d = fabsf(a - b);
    // exp(-d) in [0,1]; NEG vs NEG -> m + log(2), harmless; no NaN/Inf paths
    return m + log1pf(__expf(-d));
}

// A <- A (*) B : truncated log-space polynomial product (symmetric)
// C[j] = log sum_{a+b=j} exp(A[a] + B[b])
__device__ __forceinline__ void lpoly_mul(float* A, const float* B) {
    float out[KP1];
#pragma unroll
    for (int j = KP1 - 1; j >= 0; --j) {
        float v = A[0] + B[j];
#pragma unroll
        for (int a = 1; a < KP1; ++a) {
            if (a <= j) v = laddexp(v, A[a] + B[j - a]);
        }
        out[j] = v;
    }
#pragma unroll
    for (int j = 0; j < KP1; ++j) A[j] = out[j];
}

// ---------------- kernel: one workgroup per sample row ----------------
__global__ __launch_bounds__(TPB)
void logesp_kernel(const float* __restrict__ x, float* __restrict__ out,
                   int n_s, int n_c) {
    extern __shared__ float lds[];
    const int row  = blockIdx.x;
    const int t    = threadIdx.x;
    const float* xr = x + (size_t)row * (size_t)n_c;
    const int nvec = n_c >> 2;   // float4 groups (10000/4 = 2500, no tail here)

    // ---- stage the whole row into LDS (read HBM exactly once) ----
#if USE_ASYNC_LDS
    for (int i4 = t; i4 < nvec; i4 += TPB)
        async_copy_b128(xr + 4 * i4, lds + 4 * i4);
    for (int i = (nvec << 2) + t; i < n_c; i += TPB)   // scalar tail (generic)
        lds[i] = xr[i];
    async_wait_all();
#else
    for (int i4 = t; i4 < nvec; i4 += TPB) {
        __builtin_prefetch(xr + 4 * (i4 + 2 * TPB), 0, 0);  // global_prefetch_b8
        float4 v = ((const float4*)xr)[i4];
        ((float4*)lds)[i4] = v;
    }
    for (int i = (nvec << 2) + t; i < n_c; i += TPB)
        lds[i] = xr[i];
#endif
    __syncthreads();

    // ---- pass 1: per-thread chunk max (strided, bank-conflict-free) ----
    float m = NEGV;
    for (int i = t; i < n_c; i += TPB) m = fmaxf(m, lds[i]);

    // ---- pass 2: scaled linear-space ESP accumulation, c0 == 1 ----
    float c[KP1];
    c[0] = 1.0f;
#pragma unroll
    for (int j = 1; j < KP1; ++j) c[j] = 0.0f;
    for (int i = t; i < n_c; i += TPB) {
        float tv = __expf(lds[i] - m);   // in (0, 1]
#pragma unroll
        for (int j = KP1 - 1; j >= 1; --j)
            c[j] = fmaf(tv, c[j - 1], c[j]);
    }

    // ---- convert to log coefficients: l[j] = log c[j] + j*m ----
    float l[KP1];
    l[0] = 0.0f;
#pragma unroll
    for (int j = 1; j < KP1; ++j)
        l[j] = (c[j] > 0.0f) ? (logf(c[j]) + (float)j * m) : NEGV;

    // ---- wave32 butterfly merge (5 rounds) ----
#pragma unroll
    for (int mask = 16; mask >= 1; mask >>= 1) {
        float B[KP1];
#pragma unroll
        for (int j = 0; j < KP1; ++j) B[j] = __shfl_xor(l[j], mask, 32);
        lpoly_mul(l, B);
    }

    // ---- cross-wave merge via LDS (reuse staging buffer after barrier) ----
    __syncthreads();   // all threads done reading row data from LDS
    const int wave = t >> 5, lane = t & 31;
    if (lane == 0) {
#pragma unroll
        for (int j = 0; j < KP1; ++j) lds[wave * KP1 + j] = l[j];
    }
    __syncthreads();

    if (wave == 0) {
        const int nw = TPB / 32;   // 8 partials
        float L[KP1];
        if (lane < nw) {
#pragma unroll
            for (int j = 0; j < KP1; ++j) L[j] = lds[lane * KP1 + j];
        } else {
            L[0] = 0.0f;   // identity polynomial
#pragma unroll
            for (int j = 1; j < KP1; ++j) L[j] = NEGV;
        }
#pragma unroll
        for (int mask = 4; mask >= 1; mask >>= 1) {   // merge lanes 0..7
            float B[KP1];
#pragma unroll
            for (int j = 0; j < KP1; ++j) B[j] = __shfl_xor(L[j], mask, 32);
            lpoly_mul(L, B);
        }
        if (lane == 0) {
            out[row]        = L[4];   // log sigma_{K-1}
            out[n_s + row]  = L[5];   // log sigma_{K}
        }
    }
}

// ---------------- host-side launch ----------------
extern "C" void kernel_launch(void* const* d_in, const int* in_sizes, int n_in,
                              void* d_out, int out_size, void* d_ws, size_t ws_size,
                              hipStream_t stream) {
    (void)n_in; (void)d_ws; (void)ws_size;
    const float* x = (const float*)d_in[0];
    float* out = (float*)d_out;

    const int n_s = out_size / 2;          // 4096
    const int n_c = in_sizes[0] / n_s;     // 10000

    size_t smem = (size_t)n_c * sizeof(float);      // 40 KB row staging
    size_t smem_min = (size_t)(TPB / 32) * KP1 * sizeof(float);
    if (smem < smem_min) smem = smem_min;

    logesp_kernel<<<dim3(n_s), dim3(TPB), smem, stream>>>(x, out, n_s, n_c);
}